// ESAGEConv_7430293422225
// MI455X (gfx1250) — compile-verified
//
#include <hip/hip_runtime.h>
#include <hip/hip_bf16.h>

#define N_NODES 50000
#define N_EDGES 800000
#define D_FEAT  128

typedef __attribute__((ext_vector_type(2))) float v2f;
typedef __attribute__((ext_vector_type(8))) float v8f;

// ---------------------------------------------------------------------------
// Kernel 1: edge scatter. One wave (32 lanes) per edge; each lane moves 4
// floats (float4 gather from x[src], 4x f32 atomic add into agg[dst]).
// x (25.6MB) + agg (25.6MB) fit in the 192MB L2, so atomics resolve in L2.
// ---------------------------------------------------------------------------
__global__ void sage_scatter_kernel(const float* __restrict__ x,
                                    const long long* __restrict__ edge_index,
                                    float* __restrict__ agg,
                                    float* __restrict__ deg) {
    int tid  = blockIdx.x * blockDim.x + threadIdx.x;
    int edge = tid >> 5;
    int lane = tid & 31;
    if (edge >= N_EDGES) return;

    long long src = edge_index[edge];            // row 0: source nodes
    long long dst = edge_index[N_EDGES + edge];  // row 1: destination nodes

    const float4* xsrc = (const float4*)(x + src * D_FEAT);
    float4 v = xsrc[lane];                        // coalesced 16B/lane

    float* a = agg + dst * D_FEAT + lane * 4;
    unsafeAtomicAdd(a + 0, v.x);
    unsafeAtomicAdd(a + 1, v.y);
    unsafeAtomicAdd(a + 2, v.z);
    unsafeAtomicAdd(a + 3, v.w);

    if (lane == 0) unsafeAtomicAdd(deg + dst, 1.0f);
}

// ---------------------------------------------------------------------------
// Kernel 2: fused mean + dual GEMM + bias via V_WMMA_F32_16X16X4_F32.
//   out[16n x 16o tile] = (agg/deg) @ W_l^T + x @ W_r^T + b
// One wave per 16x16 tile; 8 waves/block cover the 8 output-feature tiles of
// the same 16 nodes. K=128 -> 32 steps of k=4, two WMMAs per step (64 total).
// A frag (16x4 f32): lane L, vgpr v -> M=L%16, K=2*(L/16)+v  (contig float2)
// B frag (4x16 f32): lane L, vgpr v -> K=2*(L/16)+v, N=L%16; B(k,n)=W[n*128+k]
// C/D  (16x16 f32): vgpr i -> M=i+8*(L/16), N=L%16
// ---------------------------------------------------------------------------
__global__ void sage_wmma_gemm_kernel(const float* __restrict__ x,
                                      const float* __restrict__ W_l,
                                      const float* __restrict__ W_r,
                                      const float* __restrict__ b,
                                      const float* __restrict__ agg,
                                      const float* __restrict__ deg,
                                      float* __restrict__ out) {
    const int lane      = threadIdx.x & 31;
    const int wave      = threadIdx.x >> 5;   // 0..7 -> output-feature tile
    const int row       = lane & 15;
    const int grp       = lane >> 4;          // 0 or 1
    const int node_base = blockIdx.x * 16;    // 3125 blocks * 16 = 50000 exact
    const int node      = node_base + row;
    const int ofeat     = wave * 16 + row;

    const float scale = 1.0f / fmaxf(deg[node], 1.0f);   // mean aggregation

    const float* aggp = agg + (size_t)node * D_FEAT;
    const float* xp   = x   + (size_t)node * D_FEAT;
    const float* wlp  = W_l + (size_t)ofeat * D_FEAT;
    const float* wrp  = W_r + (size_t)ofeat * D_FEAT;

    v8f c = {};
#pragma unroll
    for (int kk = 0; kk < D_FEAT; kk += 4) {
        const int k = kk + 2 * grp;
        v2f aa = *(const v2f*)(aggp + k);
        aa *= scale;                                   // fold mean into A frag
        v2f bl = *(const v2f*)(wlp + k);
        c = __builtin_amdgcn_wmma_f32_16x16x4_f32(
                false, aa, false, bl, (short)0, c, false, false);

        v2f ax = *(const v2f*)(xp + k);
        v2f br = *(const v2f*)(wrp + k);
        c = __builtin_amdgcn_wmma_f32_16x16x4_f32(
                false, ax, false, br, (short)0, c, false, false);
    }

    const float bias = b[ofeat];
#pragma unroll
    for (int i = 0; i < 8; ++i) {
        const int m = node_base + i + 8 * grp;
        out[(size_t)m * D_FEAT + wave * 16 + row] = c[i] + bias;
    }
}

// ---------------------------------------------------------------------------
extern "C" void kernel_launch(void* const* d_in, const int* in_sizes, int n_in,
                              void* d_out, int out_size, void* d_ws, size_t ws_size,
                              hipStream_t stream) {
    const float*     x   = (const float*)d_in[0];      // [50000,128] f32
    const float*     W_l = (const float*)d_in[1];      // [128,128]   f32
    const float*     W_r = (const float*)d_in[2];      // [128,128]   f32
    const float*     b   = (const float*)d_in[3];      // [128]       f32
    const long long* ei  = (const long long*)d_in[4];  // [2,800000]  i64

    float* agg = (float*)d_ws;                         // 50000*128 f32
    float* deg = agg + (size_t)N_NODES * D_FEAT;       // 50000     f32

    const size_t zero_bytes = ((size_t)N_NODES * D_FEAT + N_NODES) * sizeof(float);
    hipMemsetAsync(d_ws, 0, zero_bytes, stream);       // capture-safe

    {   // scatter: 32 threads per edge
        const long long total = (long long)N_EDGES * 32;
        const int block = 256;
        const int grid  = (int)((total + block - 1) / block);
        sage_scatter_kernel<<<grid, block, 0, stream>>>(x, ei, agg, deg);
    }

    {   // fused mean + GEMM: 8 waves/block, one 16-node slab per block
        const int grid = N_NODES / 16;   // 3125, exact
        sage_wmma_gemm_kernel<<<grid, 256, 0, stream>>>(
            x, W_l, W_r, b, agg, deg, (float*)d_out);
    }
}